// LightGCL_26199300505699
// MI455X (gfx1250) — compile-verified
//
#include <hip/hip_runtime.h>
#include <stdint.h>

// LightGCL forward for MI455X (gfx1250, wave32).
// SPMM: memory/atomic bound -> wave-per-nnz, float2 lanes, L2 f32 atomics.
// Rank-5 dense expansion: V_WMMA_F32_16X16X4_F32 (K=5 padded to 4+4).

#define NU 100000
#define NI 50000
#define DD 64
#define QQ 5
#define NLAYERS 2

typedef __attribute__((ext_vector_type(2))) float v2f;
typedef __attribute__((ext_vector_type(8))) float v8f;

// ---------------- streaming utilities ----------------
__global__ void k_copy_dual_f4(const float4* __restrict__ src,
                               float4* __restrict__ d1,
                               float4* __restrict__ d2, int n4) {
  int i = blockIdx.x * blockDim.x + threadIdx.x;
  if (i < n4) { float4 v = src[i]; d1[i] = v; d2[i] = v; }
}

__global__ void k_zero_f4(float4* __restrict__ p, int n4) {
  int i = blockIdx.x * blockDim.x + threadIdx.x;
  if (i < n4) p[i] = make_float4(0.f, 0.f, 0.f, 0.f);
}

__global__ void k_zero_f(float* __restrict__ p, int n) {
  int i = blockIdx.x * blockDim.x + threadIdx.x;
  if (i < n) p[i] = 0.f;
}

__global__ void k_add_f4(float4* __restrict__ a, const float4* __restrict__ b, int n4) {
  int i = blockIdx.x * blockDim.x + threadIdx.x;
  if (i < n4) {
    float4 x = a[i], y = b[i];
    x.x += y.x; x.y += y.y; x.z += y.z; x.w += y.w;
    a[i] = x;
  }
}

// ---------------- rank-5 reduction: out[5][64] = W(5xN) @ E(Nx64) ----------------
__global__ void k_lowrank_reduce(const float* __restrict__ W,
                                 const float* __restrict__ E,
                                 float* __restrict__ out, int N) {
  const int ROWS = 1024;
  int d  = threadIdx.x & 63;   // output column
  int rg = threadIdx.x >> 6;   // 0..3 row group
  int base = blockIdx.x * ROWS;
  float acc[QQ] = {0.f, 0.f, 0.f, 0.f, 0.f};
  int lim = base + ROWS; if (lim > N) lim = N;
  for (int r = base + rg; r < lim; r += 4) {
    float e = E[(size_t)r * DD + d];
    #pragma unroll
    for (int q = 0; q < QQ; ++q) acc[q] += W[(size_t)q * N + r] * e;
  }
  __shared__ float red[4][QQ][DD];
  #pragma unroll
  for (int q = 0; q < QQ; ++q) red[rg][q][d] = acc[q];
  __syncthreads();
  if (rg == 0) {
    #pragma unroll
    for (int q = 0; q < QQ; ++q) {
      float s = red[0][q][d] + red[1][q][d] + red[2][q][d] + red[3][q][d];
      atomicAdd(&out[q * DD + d], s);
    }
  }
}

// ---------------- rank-5 expansion via f32 WMMA: C(Mx64) += A(Mx5) @ B(5x64) ----------------
// One wave per 16x16 output tile. K=5 done as two 16x16x4 WMMA steps (K 0..3, K 4 + zero pad).
__global__ void k_lowrank_expand_wmma(const float* __restrict__ A,   // [M,5]
                                      const float* __restrict__ B,   // [5,64]
                                      float* __restrict__ C,         // [M,64], accumulated
                                      int mtiles) {
  int wavesPerBlock = blockDim.x >> 5;
  int wave = blockIdx.x * wavesPerBlock + (threadIdx.x >> 5);
  int total = mtiles * (DD / 16);
  if (wave >= total) return;              // wave-uniform: EXEC stays all-ones for WMMA
  int mtile = wave >> 2;                  // DD/16 == 4
  int ntile = wave & 3;
  int lane  = threadIdx.x & 31;
  int mrow  = lane & 15;
  int hi    = lane >> 4;                  // 0: K{0,1}, 1: K{2,3}
  int n     = ntile * 16 + (lane & 15);

  // A tile, ISA 16x4 f32 layout: lanes 0-15 hold K0,K1; lanes 16-31 hold K2,K3.
  const float* arow = A + (size_t)(mtile * 16 + mrow) * QQ;
  v2f a0, a1;
  a0.x = arow[hi * 2 + 0];
  a0.y = arow[hi * 2 + 1];
  a1.x = hi ? 0.f : arow[4];              // K=4 only; K=5..7 padded with zero
  a1.y = 0.f;

  // B tile, 4x16 f32 layout: VGPR v, lane L -> K = v + (L>=16 ? 2 : 0), N = L%16.
  v2f b0, b1;
  b0.x = B[(hi * 2 + 0) * DD + n];
  b0.y = B[(hi * 2 + 1) * DD + n];
  b1.x = hi ? 0.f : B[4 * DD + n];
  b1.y = 0.f;

  // C tile, 16x16 f32 layout: VGPR v, lane L -> M = v + (L>=16 ? 8 : 0), N = L%16.
  int crowBase = mtile * 16 + (hi ? 8 : 0);
  v8f c;
  #pragma unroll
  for (int v = 0; v < 8; ++v) c[v] = C[(size_t)(crowBase + v) * DD + n];

  c = __builtin_amdgcn_wmma_f32_16x16x4_f32(false, a0, false, b0, (short)0, c, false, false);
  c = __builtin_amdgcn_wmma_f32_16x16x4_f32(false, a1, false, b1, (short)0, c, false, false);

  #pragma unroll
  for (int v = 0; v < 8; ++v) C[(size_t)(crowBase + v) * DD + n] = c[v];
}

// ---------------- SPMM with inverted dropout: Z[dst] += drop(val) * E[src] ----------------
__global__ void k_spmm_atomic(const int* __restrict__ dst_idx,
                              const int* __restrict__ src_idx,
                              const float* __restrict__ vals,
                              const float* __restrict__ Esrc,
                              float* __restrict__ Zdst,
                              int nnz, unsigned seed) {
  int wavesPerBlock = blockDim.x >> 5;
  int e = blockIdx.x * wavesPerBlock + (threadIdx.x >> 5);
  if (e >= nnz) return;
  // deterministic hash Bernoulli (keep prob 0.75, inverted-dropout scaling)
  unsigned h = (unsigned)e * 0x9E3779B9u + seed;
  h ^= h >> 16; h *= 0x7FEB352Du; h ^= h >> 15; h *= 0x846CA68Bu; h ^= h >> 16;
  float u = (float)(h >> 8) * (1.0f / 16777216.0f);
  if (u >= 0.75f) return;                 // dropped (wave-uniform)
  float v = vals[e] * (1.0f / 0.75f);
  int d = dst_idx[e];
  int s = src_idx[e];
  int lane = threadIdx.x & 31;
  const float2* srow = (const float2*)(Esrc + (size_t)s * DD);
  float2 x = srow[lane];                  // 32 lanes x float2 = 64 floats, b64 loads
  float* drow = Zdst + (size_t)d * DD + lane * 2;
  atomicAdd(drow + 0, v * x.x);
  atomicAdd(drow + 1, v * x.y);
}

// ---------------- row L2 normalization (wave per 64-float row) ----------------
__global__ void k_normalize_rows(float* __restrict__ M, int nrows) {
  int wavesPerBlock = blockDim.x >> 5;
  int row = blockIdx.x * wavesPerBlock + (threadIdx.x >> 5);
  if (row >= nrows) return;
  int lane = threadIdx.x & 31;
  float2* p = (float2*)(M + (size_t)row * DD);
  float2 x = p[lane];
  float ss = x.x * x.x + x.y * x.y;
  #pragma unroll
  for (int m = 16; m; m >>= 1) ss += __shfl_xor(ss, m, 32);
  float inv = rsqrtf(ss);
  x.x *= inv; x.y *= inv;
  p[lane] = x;
}

// ---------------- host driver ----------------
extern "C" void kernel_launch(void* const* d_in, const int* in_sizes, int n_in,
                              void* d_out, int out_size, void* d_ws, size_t ws_size,
                              hipStream_t stream) {
  const float* adj_vals = (const float*)d_in[0];
  const float* E_u_0    = (const float*)d_in[1];
  const float* E_i_0    = (const float*)d_in[2];
  const float* u_mul_s  = (const float*)d_in[3];
  const float* vt       = (const float*)d_in[4];
  const float* v_mul_s  = (const float*)d_in[5];
  const float* ut       = (const float*)d_in[6];
  const int*   adj_rows = (const int*)d_in[7];
  const int*   adj_cols = (const int*)d_in[8];
  const int nnz = in_sizes[0];

  float* out    = (float*)d_out;
  float* main_u = out;                               // [NU,DD]  E_u_sum -> normalized
  float* main_i = out + (size_t)NU * DD;             // [NI,DD]
  float* aug_u  = out + (size_t)(NU + NI) * DD;      // [NU,DD]  G_u_sum
  float* aug_i  = aug_u + (size_t)NU * DD;           // [NI,DD]  G_i_sum

  float* ws = (float*)d_ws;
  float* Zu[2] = { ws, ws + (size_t)NU * DD };
  float* Zi[2] = { ws + 2 * (size_t)NU * DD,
                   ws + 2 * (size_t)NU * DD + (size_t)NI * DD };
  float* vtEi  = ws + 2 * (size_t)NU * DD + 2 * (size_t)NI * DD;  // [5][64]
  float* utEu  = vtEi + QQ * DD;                                   // [5][64]

  const int T = 256;
  const int nu4 = NU * DD / 4, ni4 = NI * DD / 4;

  // init: sums and G-sums start at E_*_0
  k_copy_dual_f4<<<(nu4 + T - 1) / T, T, 0, stream>>>((const float4*)E_u_0,
                                                      (float4*)main_u, (float4*)aug_u, nu4);
  k_copy_dual_f4<<<(ni4 + T - 1) / T, T, 0, stream>>>((const float4*)E_i_0,
                                                      (float4*)main_i, (float4*)aug_i, ni4);

  const float* Eu_cur = E_u_0;
  const float* Ei_cur = E_i_0;

  for (int layer = 0; layer < NLAYERS; ++layer) {
    // rank-5 view: s_vt_Ei = vt @ E_i, s_ut_Eu = ut @ E_u (uses pre-update E)
    k_zero_f<<<(2 * QQ * DD + T - 1) / T, T, 0, stream>>>(vtEi, 2 * QQ * DD);
    k_lowrank_reduce<<<(NI + 1023) / 1024, T, 0, stream>>>(vt, Ei_cur, vtEi, NI);
    k_lowrank_reduce<<<(NU + 1023) / 1024, T, 0, stream>>>(ut, Eu_cur, utEu, NU);

    // G_u_sum += u_mul_s @ vtEi ; G_i_sum += v_mul_s @ utEu   (WMMA f32 16x16x4)
    {
      int wavesU = (NU / 16) * (DD / 16);  // 25000
      int wavesI = (NI / 16) * (DD / 16);  // 12500
      k_lowrank_expand_wmma<<<(wavesU + 7) / 8, T, 0, stream>>>(u_mul_s, vtEi, aug_u, NU / 16);
      k_lowrank_expand_wmma<<<(wavesI + 7) / 8, T, 0, stream>>>(v_mul_s, utEu, aug_i, NI / 16);
    }

    // sparse propagation with per-layer/per-direction dropout masks
    float* ZuN = Zu[layer];
    float* ZiN = Zi[layer];
    k_zero_f4<<<(nu4 + T - 1) / T, T, 0, stream>>>((float4*)ZuN, nu4);
    k_zero_f4<<<(ni4 + T - 1) / T, T, 0, stream>>>((float4*)ZiN, ni4);
    unsigned s1 = 0xC0FFEE11u + 0x85EBCA6Bu * (unsigned)layer;
    unsigned s2 = 0xDEADBEA7u + 0x85EBCA6Bu * (unsigned)layer;
    int spmmBlocks = (nnz + 7) / 8;  // 8 waves/block, one nnz per wave
    k_spmm_atomic<<<spmmBlocks, T, 0, stream>>>(adj_rows, adj_cols, adj_vals,
                                                Ei_cur, ZuN, nnz, s1);
    k_spmm_atomic<<<spmmBlocks, T, 0, stream>>>(adj_cols, adj_rows, adj_vals,
                                                Eu_cur, ZiN, nnz, s2);

    // E_*_sum += Z_*
    k_add_f4<<<(nu4 + T - 1) / T, T, 0, stream>>>((float4*)main_u, (const float4*)ZuN, nu4);
    k_add_f4<<<(ni4 + T - 1) / T, T, 0, stream>>>((float4*)main_i, (const float4*)ZiN, ni4);

    Eu_cur = ZuN;
    Ei_cur = ZiN;
  }

  // normalize(sum/3) == normalize(sum)
  k_normalize_rows<<<(NU + 7) / 8, T, 0, stream>>>(main_u, NU);
  k_normalize_rows<<<(NI + 7) / 8, T, 0, stream>>>(main_i, NI);
}